// FR_52L_61856118997289
// MI455X (gfx1250) — compile-verified
//
#include <hip/hip_runtime.h>
#include <hip/hip_bf16.h>
#include <math.h>

// ---------------------------------------------------------------------------
// Problem constants (from reference)
// ---------------------------------------------------------------------------
#define CBIG 2800           // NF * NHB^2 * UF^2
#define NPIX 16             // 4x4 spatial pixels after time-collapse
#define NSP  112            // 7*4*4 spatial positions for the 3D convs
#define NCHUNK 11           // ceil(2800/256)

typedef float v2f __attribute__((ext_vector_type(2)));
typedef float v8f __attribute__((ext_vector_type(8)));
typedef unsigned int u32x4 __attribute__((ext_vector_type(4)));
typedef int i32x4 __attribute__((ext_vector_type(4)));
typedef int i32x8 __attribute__((ext_vector_type(8)));

// ---------------------------------------------------------------------------
// Tensor Data Mover: build a D# (ISA 8.3/8.4) and issue TENSOR_LOAD_TO_LDS.
// 2D tile: tile_d0 elements per row, tile_d1 rows, rows separated by
// row_stride elements in memory, packed densely into LDS at lds_off.
// data_size = 4 bytes.  Caller must s_wait_tensorcnt + barrier.
// This toolchain exposes the 6-arg builtin:
//   (u32x4 g0, i32x8 g1, i32x4 g2, i32x4 g3, i32x8 g4, i32 cpol)
// ---------------------------------------------------------------------------
__device__ __forceinline__ void tdm_load_to_lds(
    unsigned lds_off, const void* gsrc,
    unsigned tensor_d0, unsigned tensor_d1,
    unsigned tile_d0, unsigned tile_d1,
    unsigned long long row_stride_elems)
{
    const unsigned long long ga = (unsigned long long)(uintptr_t)gsrc;
    u32x4 g0;
    g0.x = 1u;                                           // count=1, user mode
    g0.y = lds_off;                                      // lds_addr (bytes)
    g0.z = (unsigned)(ga & 0xFFFFFFFFull);               // global_addr[31:0]
    g0.w = ((unsigned)(ga >> 32) & 0x01FFFFFFu) | (2u << 30); // [56:32] | type=2
    i32x8 g1;
    g1[0] = (int)(2u << 16);                             // wg_mask=0, data_size=4B
    g1[1] = (int)((tensor_d0 & 0xFFFFu) << 16);          // tensor_dim0[15:0]
    g1[2] = (int)(((tensor_d0 >> 16) & 0xFFFFu) | ((tensor_d1 & 0xFFFFu) << 16));
    g1[3] = (int)(((tensor_d1 >> 16) & 0xFFFFu) | ((tile_d0 & 0xFFFFu) << 16));
    g1[4] = (int)((tile_d1 & 0xFFFFu) | (1u << 16));     // tile_dim1 | tile_dim2=1
    g1[5] = (int)(unsigned)(row_stride_elems & 0xFFFFFFFFull); // dim0_stride lo
    g1[6] = (int)(unsigned)((row_stride_elems >> 32) & 0xFFFFull);
    g1[7] = 0;
    const i32x4 z4 = {0, 0, 0, 0};
    const i32x8 z8 = {0, 0, 0, 0, 0, 0, 0, 0};
    __builtin_amdgcn_tensor_load_to_lds(g0, g1, z4, z4, z8, 0);
}

__device__ __forceinline__ unsigned lds_offset_of(const void* p)
{
    // LDS aperture occupies addr[63:32]; addr[31:0] is the LDS byte offset.
    return (unsigned)(uintptr_t)p;
}

// ---------------------------------------------------------------------------
// Implicit-GEMM conv via V_WMMA_F32_16X16X4_F32, input staged in LDS by TDM.
//   A = weights [M][K] streamed from global (contiguous b64 per lane)
//   B = im2col(LDS input) [K][N], gathered from LDS (ds_load)
// Block = one N-tile (16 columns) x NWAVES M-tiles; wave w owns M-tile
// mtile0+w. All M/N are multiples of 16.  WMMA f32 fragments (ISA 7.12.2):
//   A 16x4 : lane lo=l&15 is row M=lo; VGPR v holds K = 2*(l>>4)+v
//   B 4x16 : lane lo is col N=lo;      VGPR v holds K = 2*(l>>4)+v
//   C 16x16: acc[r] = element [M = r + 8*(l>>4)][N = lo]
// ---------------------------------------------------------------------------
template<int CIN, int KD, int KH, int KW, int PD, int PH, int PW,
         int DI, int HI, int WI, int DOUT, int HO, int WO,
         int NWAVES, bool HAS_BN, bool HAS_ACT>
__global__ __launch_bounds__(NWAVES * 32)
void conv_wmma(const float* __restrict__ in,     // [CIN][DI][HI][WI]
               const float* __restrict__ wgt,    // [M][K]
               const float* __restrict__ bias,   // [M]
               const float* __restrict__ bng,    // gamma [M] (or null)
               const float* __restrict__ bnb,    // beta
               const float* __restrict__ bnm,    // mean
               const float* __restrict__ bnv,    // var
               const float* __restrict__ alp,    // prelu alpha
               float* __restrict__ out,          // [M][N]
               int M)
{
    constexpr int K     = CIN * KD * KH * KW;
    constexpr int N     = DOUT * HO * WO;
    constexpr int NT    = N / 16;
    constexpr int NELEM = CIN * DI * HI * WI;
    constexpr bool KTAIL = (K & 3) != 0;

    extern __shared__ float smemIn[];            // NELEM floats

    // ---- Stage the whole input tensor into LDS with the TDM (wave 0 issues;
    //      TENSORcnt waited by the issuing wave; barrier publishes to block).
    if (threadIdx.x == 0) {
        tdm_load_to_lds(lds_offset_of(smemIn), in,
                        NELEM, 1, NELEM, 1, (unsigned long long)NELEM);
        __builtin_amdgcn_s_wait_tensorcnt(0);
    }
    __syncthreads();

    const int nt     = blockIdx.x % NT;
    const int mtile0 = (blockIdx.x / NT) * NWAVES;
    const int wave   = threadIdx.x >> 5;
    const int lane   = threadIdx.x & 31;
    const int lo     = lane & 15;
    const int hi     = lane >> 4;
    const int mt     = mtile0 + wave;
    if (mt * 16 >= M) return;                    // whole-wave uniform guard
    const int m0 = mt * 16, n0 = nt * 16;

    // Output spatial coords for this lane's B column (constant over K loop)
    const int n   = n0 + lo;
    const int od  = n / (HO * WO);
    const int ohw = n % (HO * WO);
    const int oh  = ohw / WO;
    const int ow_ = ohw % WO;

    const float* __restrict__ arow = wgt + (size_t)(m0 + lo) * K;

    v8f acc = {};
    for (int k0 = 0; k0 < K; k0 += 4) {
        const int ka = k0 + 2 * hi;
        v2f a, b;
#pragma unroll
        for (int v = 0; v < 2; ++v) {
            const int k = ka + v;
            // ---- A fragment (weights, row-major contiguous in K)
            float av;
            if constexpr (KTAIL) av = (k < K) ? arow[k] : 0.0f;
            else                 av = arow[k];
            a[v] = av;
            // ---- B fragment (im2col gather from LDS, zero padding)
            float bv = 0.0f;
            bool kok = true;
            if constexpr (KTAIL) kok = (k < K);
            if (kok) {
                const int c  = k / (KD * KH * KW);
                const int r1 = k % (KD * KH * KW);
                const int kd = r1 / (KH * KW);
                const int r2 = r1 % (KH * KW);
                const int kh = r2 / KW;
                const int kw = r2 % KW;
                const int id = od + kd - PD;
                const int ih = oh + kh - PH;
                const int iw = ow_ + kw - PW;
                if (id >= 0 && id < DI && ih >= 0 && ih < HI && iw >= 0 && iw < WI)
                    bv = smemIn[((c * DI + id) * HI + ih) * WI + iw];
            }
            b[v] = bv;
        }
        acc = __builtin_amdgcn_wmma_f32_16x16x4_f32(
            /*neg_a=*/false, a, /*neg_b=*/false, b,
            /*c_mod=*/(short)0, acc, /*reuse_a=*/false, /*reuse_b=*/false);
    }

    // Epilogue: bias (+BN +PReLU), write [M][N]
#pragma unroll
    for (int r = 0; r < 8; ++r) {
        const int m = m0 + r + 8 * hi;
        float v = acc[r] + bias[m];
        if constexpr (HAS_BN) {
            const float sc = bng[m] * rsqrtf(bnv[m] + 1e-5f);
            v = (v - bnm[m]) * sc + bnb[m];
        }
        if constexpr (HAS_ACT) {
            v = (v >= 0.0f) ? v : alp[m] * v;
        }
        out[(size_t)m * N + n] = v;
    }
}

// ---------------------------------------------------------------------------
// Attention part.  o is [CBIG][NPIX] (channel-major).  Per pixel p:
//   m_i = max_j(o_i*o_j) = max(o_i*omax, o_i*omin)
//   S_i = sum_j exp(o_i*o_j - m_i);   w_i = o_i / S_i
//   r_j = sum_i w_i * exp(o_i*o_j - m_i)
//   out_j = softmax_j(r)
// Column loads (stride NPIX) use TDM 2D gathers into LDS.
// ---------------------------------------------------------------------------
__global__ __launch_bounds__(256)
void att_minmax(const float* __restrict__ o, float* __restrict__ omax,
                float* __restrict__ omin)
{
    __shared__ float smx[256], smn[256];
    const int p = blockIdx.x, t = threadIdx.x;
    float mx = -3.4e38f, mn = 3.4e38f;
    for (int i = t; i < CBIG; i += 256) {
        const float v = o[i * NPIX + p];
        mx = fmaxf(mx, v);
        mn = fminf(mn, v);
    }
    smx[t] = mx; smn[t] = mn;
    __syncthreads();
    for (int s = 128; s > 0; s >>= 1) {
        if (t < s) {
            smx[t] = fmaxf(smx[t], smx[t + s]);
            smn[t] = fminf(smn[t], smn[t + s]);
        }
        __syncthreads();
    }
    if (t == 0) { omax[p] = smx[0]; omin[p] = smn[0]; }
}

__global__ __launch_bounds__(256)
void att_rowstats(const float* __restrict__ o, const float* __restrict__ omax,
                  const float* __restrict__ omin, float* __restrict__ marr,
                  float* __restrict__ warr)
{
    __shared__ float so[CBIG];
    const int p = blockIdx.x / NCHUNK;
    const int chunk = blockIdx.x % NCHUNK;
    const int t = threadIdx.x;
    if (t == 0) {
        // column p of o: 2800 rows of 1 element, row stride NPIX
        tdm_load_to_lds(lds_offset_of(so), o + p, NPIX, CBIG, 1, CBIG, NPIX);
        __builtin_amdgcn_s_wait_tensorcnt(0);
    }
    __syncthreads();
    const int i = chunk * 256 + t;
    if (i < CBIG) {
        const float oi = so[i];
        const float m  = fmaxf(oi * omax[p], oi * omin[p]);
        float S = 0.0f;
        for (int j = 0; j < CBIG; ++j) S += __expf(oi * so[j] - m);
        marr[i * NPIX + p] = m;
        warr[i * NPIX + p] = oi / S;
    }
}

__global__ __launch_bounds__(256)
void att_accum(const float* __restrict__ o, const float* __restrict__ marr,
               const float* __restrict__ warr, float* __restrict__ rarr)
{
    __shared__ float so[CBIG], sm[CBIG], sw[CBIG];
    const int p = blockIdx.x / NCHUNK;
    const int chunk = blockIdx.x % NCHUNK;
    const int t = threadIdx.x;
    if (t == 0) {
        tdm_load_to_lds(lds_offset_of(so), o    + p, NPIX, CBIG, 1, CBIG, NPIX);
        tdm_load_to_lds(lds_offset_of(sm), marr + p, NPIX, CBIG, 1, CBIG, NPIX);
        tdm_load_to_lds(lds_offset_of(sw), warr + p, NPIX, CBIG, 1, CBIG, NPIX);
        __builtin_amdgcn_s_wait_tensorcnt(0);
    }
    __syncthreads();
    const int j = chunk * 256 + t;
    if (j < CBIG) {
        const float oj = so[j];
        float acc = 0.0f;
        for (int i = 0; i < CBIG; ++i) acc += sw[i] * __expf(so[i] * oj - sm[i]);
        rarr[j * NPIX + p] = acc;
    }
}

__global__ __launch_bounds__(256)
void att_softmax(const float* __restrict__ rarr, float* __restrict__ out)
{
    __shared__ float sr[CBIG];
    __shared__ float red[256];
    const int p = blockIdx.x, t = threadIdx.x;
    if (t == 0) {
        tdm_load_to_lds(lds_offset_of(sr), rarr + p, NPIX, CBIG, 1, CBIG, NPIX);
        __builtin_amdgcn_s_wait_tensorcnt(0);
    }
    __syncthreads();
    float mx = -3.4e38f;
    for (int j = t; j < CBIG; j += 256) mx = fmaxf(mx, sr[j]);
    red[t] = mx;
    __syncthreads();
    for (int s = 128; s > 0; s >>= 1) {
        if (t < s) red[t] = fmaxf(red[t], red[t + s]);
        __syncthreads();
    }
    mx = red[0];
    __syncthreads();
    float sum = 0.0f;
    for (int j = t; j < CBIG; j += 256) {
        const float e = __expf(sr[j] - mx);
        sr[j] = e;
        sum += e;
    }
    red[t] = sum;
    __syncthreads();
    for (int s = 128; s > 0; s >>= 1) {
        if (t < s) red[t] += red[t + s];
        __syncthreads();
    }
    const float inv = 1.0f / red[0];
    for (int j = t; j < CBIG; j += 256) out[j * NPIX + p] = sr[j] * inv;
}

// ---------------------------------------------------------------------------
// Host launcher.
// Input order (setup_inputs dict flattened):
//   0: x
//   1..14:  pre: w1,b1,bn1(g,b,m,v),a1,w2,b2,bn2(g,b,m,v),a2
//   15+18*s: fw1,fb1,fbn1(4),fa1,fw2,fb2,fbn2(4),fa2,ow1,ob1,ow2,ob2
// ---------------------------------------------------------------------------
extern "C" void kernel_launch(void* const* d_in, const int* in_sizes, int n_in,
                              void* d_out, int out_size, void* d_ws, size_t ws_size,
                              hipStream_t stream)
{
    (void)in_sizes; (void)n_in; (void)out_size; (void)ws_size;
    auto F = [&](int i) { return (const float*)d_in[i]; };

    const float* x = F(0);

    // Workspace carve-up (floats)
    float* ws   = (float*)d_ws;
    float* h    = ws;                 // 128*112 = 14336
    float* feat = h + 14336;          // 288*112 = 32256  (x0 in first 256 ch)
    float* f1   = feat + 32256;       // 128*112 = 14336
    float* o1   = f1 + 14336;         // 32*16   = 512
    float* o2   = o1 + 512;           // 2800*16 = 44800
    float* omax = o2 + 44800;         // 16
    float* omin = omax + 16;          // 16
    float* marr = omin + 16;          // 44800
    float* warr = marr + 44800;       // 44800
    float* rarr = warr + 44800;       // 44800

    float* featF = feat + 256 * NSP;  // fw2 output region (32 ch) of concat

    // ---- pre conv1: (1,5,5) pad (0,2,2), 3->128, BN+PReLU : x -> h
    conv_wmma<3,1,5,5, 0,2,2, 7,4,4, 7,4,4, 8, true,true>
        <<<7, 8*32, 3*NSP*4, stream>>>(
            x, F(1), F(2), F(3), F(4), F(5), F(6), F(7), h, 128);

    // ---- pre conv2: (3,1,1) pad (1,0,0), 128->256, BN+PReLU : h -> feat[0:256]
    conv_wmma<128,3,1,1, 1,0,0, 7,4,4, 7,4,4, 16, true,true>
        <<<7, 16*32, 128*NSP*4, stream>>>(
            h, F(8), F(9), F(10), F(11), F(12), F(13), F(14), feat, 256);

    float* out = (float*)d_out;
    for (int s = 0; s < 4; ++s) {
        const int b = 15 + 18 * s;
        // fw1: 3x3x3 pad 1, in_c -> 128, BN+PReLU : feat -> f1
        if (s == 0) {
            conv_wmma<256,3,3,3, 1,1,1, 7,4,4, 7,4,4, 8, true,true>
                <<<7, 8*32, 256*NSP*4, stream>>>(
                    feat, F(b+0), F(b+1), F(b+2), F(b+3), F(b+4), F(b+5), F(b+6),
                    f1, 128);
        } else {
            conv_wmma<288,3,3,3, 1,1,1, 7,4,4, 7,4,4, 8, true,true>
                <<<7, 8*32, 288*NSP*4, stream>>>(
                    feat, F(b+0), F(b+1), F(b+2), F(b+3), F(b+4), F(b+5), F(b+6),
                    f1, 128);
        }
        // fw2: 3x3x3 pad 1, 128 -> 32, BN+PReLU : f1 -> feat[256:288] (concat)
        conv_wmma<128,3,3,3, 1,1,1, 7,4,4, 7,4,4, 2, true,true>
            <<<7, 2*32, 128*NSP*4, stream>>>(
                f1, F(b+7), F(b+8), F(b+9), F(b+10), F(b+11), F(b+12), F(b+13),
                featF, 32);
        // ow1: (7,1,1) valid, 32 -> 32 : featF -> o1  [32][16]
        conv_wmma<32,7,1,1, 0,0,0, 7,4,4, 1,4,4, 2, false,false>
            <<<1, 2*32, 32*NSP*4, stream>>>(
                featF, F(b+14), F(b+15), nullptr, nullptr, nullptr, nullptr,
                nullptr, o1, 32);
        // ow2: 1x1x1, 32 -> 2800 : o1 -> o2  [2800][16]
        conv_wmma<32,1,1,1, 0,0,0, 1,4,4, 1,4,4, 16, false,false>
            <<<11, 16*32, 32*16*4, stream>>>(
                o1, F(b+16), F(b+17), nullptr, nullptr, nullptr, nullptr,
                nullptr, o2, 2800);

        // attention + softmax
        att_minmax  <<<NPIX, 256, 0, stream>>>(o2, omax, omin);
        att_rowstats<<<NPIX * NCHUNK, 256, 0, stream>>>(o2, omax, omin, marr, warr);
        att_accum   <<<NPIX * NCHUNK, 256, 0, stream>>>(o2, marr, warr, rarr);
        att_softmax <<<NPIX, 256, 0, stream>>>(rarr, out + (size_t)s * CBIG * NPIX);
    }
}